// GConv_44521630991152
// MI455X (gfx1250) — compile-verified
//
#include <hip/hip_runtime.h>

#define N_NODES 100000
#define N_EDGES 600000
#define F_IN 128
#define F_OUT 128

typedef __attribute__((ext_vector_type(2))) float v2f;
typedef __attribute__((ext_vector_type(8))) float v8f;

// ---------------------------------------------------------------------------
// Kernel 1: out[n, f] = bias[f]   (atomic scatter accumulates on top)
// ---------------------------------------------------------------------------
__global__ __launch_bounds__(256) void gconv_init_bias(
    const float* __restrict__ bias, float* __restrict__ out, int total) {
  int i = blockIdx.x * blockDim.x + threadIdx.x;
  if (i < total) out[i] = bias[i & (F_OUT - 1)];
}

// ---------------------------------------------------------------------------
// Kernel 2: support = X @ W via V_WMMA_F32_16X16X4_F32 (fp32 in, fp32 acc).
// One wave per 16x16 output tile; 8 waves/block cover the 8 column tiles of
// F_OUT=128; grid.x covers 100000/16 = 6250 row tiles. K-loop: 128/4 = 32
// WMMA ops, accumulator chained through C.
//
// VGPR layouts per CDNA5 ISA 7.12.2:
//  A (16x4 f32):  lanes 0-15 hold M=lane, VGPR0=K0, VGPR1=K1;
//                 lanes 16-31 hold M=lane-16, VGPR0=K2, VGPR1=K3.
//  B (4x16 f32):  VGPR0: lanes0-15 = row K0 (N=lane), lanes16-31 = row K2;
//                 VGPR1: rows K1 / K3.   (C/D-style half-lane split)
//  C/D (16x16):   VGPR r: lanes0-15 -> M=r, lanes16-31 -> M=r+8, N=lane&15.
// ---------------------------------------------------------------------------
__global__ __launch_bounds__(256) void gconv_gemm_wmma(
    const float* __restrict__ X, const float* __restrict__ W,
    float* __restrict__ S) {
  const int lane = threadIdx.x & 31;
  const int wave = threadIdx.x >> 5;   // 0..7 -> output column tile
  const int m0   = blockIdx.x * 16;
  const int n0   = wave * 16;
  const int half = lane >> 4;          // 0: lanes 0-15, 1: lanes 16-31
  const int l15  = lane & 15;

  const float* xrow = X + (size_t)(m0 + l15) * F_IN;  // A row for this lane
  const int    bcol = n0 + l15;                       // B/N column for this lane

  v8f acc = {};
#pragma unroll
  for (int k = 0; k < F_IN; k += 4) {
    // A: two consecutive K values (8-byte aligned: k%4==0, half*2 even)
    v2f a = *(const v2f*)(xrow + k + half * 2);
    // B: rows k+half*2 and k+half*2+1 of W, column bcol
    v2f b;
    b.x = W[(size_t)(k + half * 2 + 0) * F_OUT + bcol];
    b.y = W[(size_t)(k + half * 2 + 1) * F_OUT + bcol];
    acc = __builtin_amdgcn_wmma_f32_16x16x4_f32(
        /*neg_a=*/false, a, /*neg_b=*/false, b,
        /*c_mod=*/(short)0, acc, /*reuse_a=*/false, /*reuse_b=*/false);
  }

#pragma unroll
  for (int r = 0; r < 8; ++r) {
    int row = m0 + r + half * 8;
    S[(size_t)row * F_OUT + bcol] = acc[r];
  }
}

// ---------------------------------------------------------------------------
// Kernel 3: COO SpMM scatter. One wave per edge: lane L moves float4 at
// feature offset 4*L (32 lanes x 4 = 128 floats), scales by val, and
// accumulates into out[row] with hardware global_atomic_add_f32.
// support & out are L2-resident (51.2 MB each << 192 MB L2).
// ---------------------------------------------------------------------------
__global__ __launch_bounds__(256) void gconv_spmm_scatter(
    const float* __restrict__ support, const int* __restrict__ rows,
    const int* __restrict__ cols, const float* __restrict__ vals,
    float* __restrict__ out, int n_edges) {
  int gwave = (blockIdx.x * blockDim.x + threadIdx.x) >> 5;
  int lane  = threadIdx.x & 31;
  if (gwave >= n_edges) return;

  int   r = rows[gwave];
  int   c = cols[gwave];
  float v = vals[gwave];

  const float4* src = (const float4*)(support + (size_t)c * F_OUT);
  float4 m = src[lane];

  float* dst = out + (size_t)r * F_OUT + lane * 4;
  unsafeAtomicAdd(dst + 0, m.x * v);
  unsafeAtomicAdd(dst + 1, m.y * v);
  unsafeAtomicAdd(dst + 2, m.z * v);
  unsafeAtomicAdd(dst + 3, m.w * v);
}

// ---------------------------------------------------------------------------
// Launch: init(out=bias) -> GEMM(support) -> scatter(edges0) -> scatter(edges1)
// All on `stream`; support lives in d_ws (needs 100000*128*4 = 51.2 MB).
// ---------------------------------------------------------------------------
extern "C" void kernel_launch(void* const* d_in, const int* in_sizes, int n_in,
                              void* d_out, int out_size, void* d_ws,
                              size_t ws_size, hipStream_t stream) {
  const float* x     = (const float*)d_in[0];
  const float* w     = (const float*)d_in[1];
  const float* bias  = (const float*)d_in[2];
  const float* vals0 = (const float*)d_in[3];
  const float* vals1 = (const float*)d_in[4];
  const int*   rows0 = (const int*)d_in[5];
  const int*   cols0 = (const int*)d_in[6];
  const int*   rows1 = (const int*)d_in[7];
  const int*   cols1 = (const int*)d_in[8];

  float* out     = (float*)d_out;
  float* support = (float*)d_ws;

  const int total = N_NODES * F_OUT;
  gconv_init_bias<<<(total + 255) / 256, 256, 0, stream>>>(bias, out, total);

  gconv_gemm_wmma<<<N_NODES / 16, 256, 0, stream>>>(x, w, support);

  const int edges_per_block = 256 / 32;
  const int sgrid = (N_EDGES + edges_per_block - 1) / edges_per_block;
  gconv_spmm_scatter<<<sgrid, 256, 0, stream>>>(support, rows0, cols0, vals0,
                                                out, N_EDGES);
  gconv_spmm_scatter<<<sgrid, 256, 0, stream>>>(support, rows1, cols1, vals1,
                                                out, N_EDGES);
}